// MpMG2P_34041910788795
// MI455X (gfx1250) — compile-verified
//
#include <hip/hip_runtime.h>

// MPM G2P gather: out[p] = sum_{a,b,c} wx[a]*wy[b]*wz[c] * grid[base+(a,b,c)]
// Grid 144^3 x float3 (35.8 MB) is L2-resident on MI455X (192 MB L2).
// pos/out are streamed with non-temporal hints so they do not evict the grid.

#define N_GRID 144

__global__ __launch_bounds__(256) void g2p_gather_kernel(
    const float* __restrict__ grid,   // (144,144,144,3) f32
    const float* __restrict__ pos,    // (N,3) f32
    float* __restrict__ out,          // (N,3) f32
    int n)
{
    const int i = blockIdx.x * blockDim.x + threadIdx.x;
    if (i >= n) return;

    // --- streaming pos load, non-temporal (keep L2 for the grid) ---
    const float px = __builtin_nontemporal_load(pos + 3 * i + 0);
    const float py = __builtin_nontemporal_load(pos + 3 * i + 1);
    const float pz = __builtin_nontemporal_load(pos + 3 * i + 2);

    // npos = pos / (1/128) == pos * 128 (exact)
    const float nx = px * 128.0f;
    const float ny = py * 128.0f;
    const float nz = pz * 128.0f;

    const int ix = (int)nx;   // pos in [0,1) -> base in [0,127]
    const int iy = (int)ny;
    const int iz = (int)nz;

    const float lx = nx - (float)ix;
    const float ly = ny - (float)iy;
    const float lz = nz - (float)iz;

    // quadratic B-spline weights per axis
    float wx[3], wy[3], wz[3];
    {
        const float ax = 1.0f - lx, bx = 0.5f - lx;
        wx[0] = 0.5f * ax * ax; wx[1] = 0.75f - bx * bx; wx[2] = 0.5f * lx * lx;
        const float ay = 1.0f - ly, by = 0.5f - ly;
        wy[0] = 0.5f * ay * ay; wy[1] = 0.75f - by * by; wy[2] = 0.5f * ly * ly;
        const float az = 1.0f - lz, bz = 0.5f - lz;
        wz[0] = 0.5f * az * az; wz[1] = 0.75f - bz * bz; wz[2] = 0.5f * lz * lz;
    }

    const int strideY = N_GRID * 3;            // floats per y-line
    const int strideX = N_GRID * N_GRID * 3;   // floats per x-plane
    const float* base = grid + ix * strideX + iy * strideY + iz * 3;

    float ox = 0.0f, oy = 0.0f, oz = 0.0f;

    // 9 rows of 9 contiguous floats each (z-neighbors are contiguous in memory).
    // Loads within a row are mergeable into b96/b128 vector loads; grid loads
    // use default (RT) cache policy so they stay resident in the 192MB L2.
#pragma unroll
    for (int a = 0; a < 3; ++a) {
#pragma unroll
        for (int b = 0; b < 3; ++b) {
            const float* __restrict__ rp = base + a * strideX + b * strideY;
            float r0 = rp[0], r1 = rp[1], r2 = rp[2];
            float r3 = rp[3], r4 = rp[4], r5 = rp[5];
            float r6 = rp[6], r7 = rp[7], r8 = rp[8];

            // reduce over c with wz, then scale once by wx[a]*wy[b]
            const float sx = fmaf(wz[0], r0, fmaf(wz[1], r3, wz[2] * r6));
            const float sy = fmaf(wz[0], r1, fmaf(wz[1], r4, wz[2] * r7));
            const float sz = fmaf(wz[0], r2, fmaf(wz[1], r5, wz[2] * r8));

            const float wab = wx[a] * wy[b];
            ox = fmaf(wab, sx, ox);
            oy = fmaf(wab, sy, oy);
            oz = fmaf(wab, sz, oz);
        }
    }

    // --- streaming out store, non-temporal ---
    __builtin_nontemporal_store(ox, out + 3 * i + 0);
    __builtin_nontemporal_store(oy, out + 3 * i + 1);
    __builtin_nontemporal_store(oz, out + 3 * i + 2);
}

extern "C" void kernel_launch(void* const* d_in, const int* in_sizes, int n_in,
                              void* d_out, int out_size, void* d_ws, size_t ws_size,
                              hipStream_t stream) {
    const float* grid = (const float*)d_in[0];  // (144,144,144,3)
    const float* pos  = (const float*)d_in[1];  // (N,3)
    float* out        = (float*)d_out;          // (N,3)

    const int n = in_sizes[1] / 3;              // particle count
    const int block = 256;                      // 8 wave32s per block
    const int gridDimX = (n + block - 1) / block;

    g2p_gather_kernel<<<gridDimX, block, 0, stream>>>(grid, pos, out, n);
}